// ChebGraphConvolution_88055419503321
// MI455X (gfx1250) — compile-verified
//
#include <hip/hip_runtime.h>
#include <hip/hip_bf16.h>

typedef __attribute__((ext_vector_type(2))) float v2f;
typedef __attribute__((ext_vector_type(8))) float v8f;

#define DFEAT 96
#define NTILES_N 6  // 96/16

// ---------------------------------------------------------------------------
// SpMM scatter: one wave (32 lanes) per edge, 3 features per lane.
// Y[rows[e], :] += vals[e] * X[cols[e], :]
// ---------------------------------------------------------------------------
__global__ void cheb_scatter96(const int* __restrict__ rows,
                               const int* __restrict__ cols,
                               const float* __restrict__ vals,
                               const float* __restrict__ X,
                               float* __restrict__ Y,
                               int nEdges) {
    int t = blockIdx.x * blockDim.x + threadIdx.x;
    int e = t >> 5;          // edge index (wave-uniform)
    if (e >= nEdges) return;
    int lane = t & 31;
    int r = rows[e];
    int c = cols[e];
    float v = vals[e];
    const float* xr = X + (size_t)c * DFEAT;
    float*       yr = Y + (size_t)r * DFEAT;
#pragma unroll
    for (int j = lane; j < DFEAT; j += 32) {
        unsafeAtomicAdd(&yr[j], v * xr[j]);   // -> global_atomic_add_f32
    }
}

// ---------------------------------------------------------------------------
// k=1 combine: T1 = 2*Y - H ; S = H + T1 ; Y = 0
// ---------------------------------------------------------------------------
__global__ void cheb_combine_k1(const float* __restrict__ H,
                                float* __restrict__ Y,
                                float* __restrict__ T1,
                                float* __restrict__ S,
                                int n) {
    int i = blockIdx.x * blockDim.x + threadIdx.x;
    if (i >= n) return;
    float h = H[i];
    float y = Y[i];
    float t = 2.0f * y - h;
    T1[i] = t;
    S[i]  = h + t;
    Y[i]  = 0.0f;
}

// ---------------------------------------------------------------------------
// k>=2 combine: Tnext = 4*Y - 2*Tcur - Tprev ; S += Tnext ; optionally Y = 0
// (Tnext may alias Tprev: we read Tprev[i] before writing Tnext[i], same i.)
// ---------------------------------------------------------------------------
__global__ void cheb_combine_rec(const float* Tprev,
                                 const float* __restrict__ Tcur,
                                 float* __restrict__ Y,
                                 float* Tnext,
                                 float* __restrict__ S,
                                 int n, int zeroY) {
    int i = blockIdx.x * blockDim.x + threadIdx.x;
    if (i >= n) return;
    float t = 4.0f * Y[i] - 2.0f * Tcur[i] - Tprev[i];
    Tnext[i] = t;
    S[i] += t;
    if (zeroY) Y[i] = 0.0f;
}

// ---------------------------------------------------------------------------
// Dense GEMM: out[50000x96] = S[50000x96] @ W[96x96] + bias, fp32 WMMA.
// One wave32 per 16x16 output tile; K-loop of 24 x V_WMMA_F32_16X16X4_F32.
//
// A 16x4 f32 layout:   lanes 0-15 M=0..15 {K=0,K=1}; lanes 16-31 {K=2,K=3}
// B 4x16 f32 layout:   lanes 0-15 N=0..15 {K=0,K=1}; lanes 16-31 {K=2,K=3}
// C/D 16x16 f32:       VGPR r: lanes 0-15 -> M=r, N=lane;
//                              lanes 16-31 -> M=8+r, N=lane-16
// ---------------------------------------------------------------------------
__global__ void cheb_gemm_wmma(const float* __restrict__ S,
                               const float* __restrict__ W,
                               const float* __restrict__ bias,
                               float* __restrict__ out,
                               int nTilesM) {
    int waveId = blockIdx.x * (blockDim.x >> 5) + (threadIdx.x >> 5);
    int nTiles = nTilesM * NTILES_N;
    if (waveId >= nTiles) return;               // wave-uniform: EXEC stays full

    int mt = waveId / NTILES_N;
    int nt = waveId - mt * NTILES_N;
    int lane = threadIdx.x & 31;
    int half = lane >> 4;                        // 0: K pair {0,1}, 1: {2,3}
    int l    = lane & 15;

    const float* Srow = S + (size_t)(mt * 16 + l) * DFEAT;
    const float* Wcol = W + nt * 16 + l;

    v8f c = {};
#pragma unroll
    for (int k0 = 0; k0 < DFEAT; k0 += 4) {
        int k = k0 + 2 * half;
        v2f a, b;
        a.x = Srow[k];
        a.y = Srow[k + 1];
        b.x = Wcol[(size_t)k * DFEAT];
        b.y = Wcol[(size_t)(k + 1) * DFEAT];
        c = __builtin_amdgcn_wmma_f32_16x16x4_f32(
                /*neg_a=*/false, a, /*neg_b=*/false, b,
                /*c_mod=*/(short)0, c, /*reuse_a=*/false, /*reuse_b=*/false);
    }

    int n  = nt * 16 + l;
    float bv = bias[n];
    int mbase = mt * 16 + half * 8;
    float* orow = out + (size_t)mbase * DFEAT + n;
#pragma unroll
    for (int r = 0; r < 8; ++r) {
        orow[(size_t)r * DFEAT] = c[r] + bv;
    }
}

// ---------------------------------------------------------------------------
// Host-side orchestration
// ---------------------------------------------------------------------------
extern "C" void kernel_launch(void* const* d_in, const int* in_sizes, int n_in,
                              void* d_out, int out_size, void* d_ws, size_t ws_size,
                              hipStream_t stream) {
    const int*   rows = (const int*)d_in[0];
    const int*   cols = (const int*)d_in[1];
    const float* vals = (const float*)d_in[2];
    const float* H    = (const float*)d_in[3];
    const float* W    = (const float*)d_in[4];
    const float* bias = (const float*)d_in[5];
    float* out = (float*)d_out;

    const int nEdges = in_sizes[0];
    const int nNodes = in_sizes[3] / DFEAT;
    const int nElem  = nNodes * DFEAT;

    // Workspace layout: Y | S | bufA | bufB  (each nElem floats)
    float* Y    = (float*)d_ws;
    float* Ssum = Y + nElem;
    float* bufA = Ssum + nElem;   // holds T2
    float* bufB = bufA + nElem;   // holds T1, then T3

    const int TPB = 256;
    dim3 blk(TPB);
    dim3 grdScatter((nEdges * 32 + TPB - 1) / TPB);
    dim3 grdElem((nElem + TPB - 1) / TPB);
    const int nTilesM = nNodes / 16;             // 50000/16 = 3125
    int nWaves = nTilesM * NTILES_N;
    dim3 grdGemm((nWaves + (TPB / 32) - 1) / (TPB / 32));

    // Y = 0
    hipMemsetAsync(Y, 0, (size_t)nElem * sizeof(float), stream);

    // ---- k = 1: T1 = 2*spmm(H) - H ; S = H + T1
    cheb_scatter96<<<grdScatter, blk, 0, stream>>>(rows, cols, vals, H, Y, nEdges);
    cheb_combine_k1<<<grdElem, blk, 0, stream>>>(H, Y, bufB, Ssum, nElem);

    // ---- k = 2: T2 = 4*spmm(T1) - 2*T1 - H ; S += T2
    cheb_scatter96<<<grdScatter, blk, 0, stream>>>(rows, cols, vals, bufB, Y, nEdges);
    cheb_combine_rec<<<grdElem, blk, 0, stream>>>(H, bufB, Y, bufA, Ssum, nElem, 1);

    // ---- k = 3: T3 = 4*spmm(T2) - 2*T2 - T1 ; S += T3   (T3 overwrites T1)
    cheb_scatter96<<<grdScatter, blk, 0, stream>>>(rows, cols, vals, bufA, Y, nEdges);
    cheb_combine_rec<<<grdElem, blk, 0, stream>>>(bufB, bufA, Y, bufB, Ssum, nElem, 0);

    // ---- out = S @ W + bias (fp32 WMMA)
    cheb_gemm_wmma<<<grdGemm, blk, 0, stream>>>(Ssum, W, bias, out, nTilesM);
}